// Generator3DLUTL5_identity_32693291057265
// MI455X (gfx1250) — compile-verified
//
#include <hip/hip_runtime.h>
#include <hip/hip_bf16.h>
#include <stdint.h>

// Quadrilinear 4D LUT interpolation for MI455X (gfx1250).
//   x:   (B=5, 4, 1024, 1024) f32 coords in [0,1]
//   LUT: (B=5, 1, 9,9,9,9)    f32
//   out: (B=5, 1, 1024, 1024) f32
// Strategy: per-batch 26 KB LUT staged into LDS with gfx1250 async
// global->LDS copies (ASYNCcnt), then bandwidth-bound streaming of the
// coordinate planes with 16 LDS gathers per pixel.

#define LUT_D    9
#define LUT_N    (LUT_D * LUT_D * LUT_D * LUT_D)   // 6561
#define LUT_PAD  6656                              // 26 * 256 (loop-friendly pad)
#define HW       (1024 * 1024)
#define NBATCH   5
#define NBLK     256                               // blocks per batch
#define TPB      256                               // 8 wave32 waves

__global__ __launch_bounds__(TPB) void lut4d_interp_kernel(
    const float* __restrict__ x,
    const float* __restrict__ lut,
    float* __restrict__ out)
{
    __shared__ float sLUT[LUT_PAD];

    const int b   = blockIdx.y;
    const int tid = threadIdx.x;

    // ---- Stage LUT[b] into LDS via gfx1250 async global->LDS DMA ----
    {
        const uint64_t gbase = (uint64_t)(uintptr_t)(lut + (size_t)b * LUT_N);
        for (int i = 0; i < LUT_PAD; i += TPB) {
            const int slot = i + tid;                        // LDS slot (padded)
            const int j    = slot < LUT_N ? slot : (LUT_N - 1); // clamp global idx
            // Low 32 bits of the flat address of a __shared__ object are the
            // LDS byte offset (aperture id lives in the high dword).
            unsigned lds_addr = (unsigned)(uintptr_t)(&sLUT[slot]);
            unsigned goff     = (unsigned)(j * 4);
            asm volatile("global_load_async_to_lds_b32 %0, %1, %2"
                         :
                         : "v"(lds_addr), "v"(goff), "s"(gbase)
                         : "memory");
        }
        // Drain this wave's async copies, then barrier so every wave sees LDS.
        asm volatile("s_wait_asynccnt 0" ::: "memory");
    }
    __syncthreads();

    // ---- Streaming interpolation ----
    const float4* __restrict__ X0 = (const float4*)(x + (size_t)(b * 4 + 0) * HW);
    const float4* __restrict__ X1 = (const float4*)(x + (size_t)(b * 4 + 1) * HW);
    const float4* __restrict__ X2 = (const float4*)(x + (size_t)(b * 4 + 2) * HW);
    const float4* __restrict__ X3 = (const float4*)(x + (size_t)(b * 4 + 3) * HW);
    float4* __restrict__ O        = (float4*)(out + (size_t)b * HW);

    const int vecPerBatch = HW / 4;              // float4 elements per batch plane
    const int vecPerBlk   = vecPerBatch / NBLK;  // 1024
    const int vbase       = blockIdx.x * vecPerBlk;

    auto interp1 = [&](float c0, float c1, float c2, float c3) -> float {
        float t0 = fminf(fmaxf(c0, 0.0f), 1.0f) * (float)(LUT_D - 1);
        float t1 = fminf(fmaxf(c1, 0.0f), 1.0f) * (float)(LUT_D - 1);
        float t2 = fminf(fmaxf(c2, 0.0f), 1.0f) * (float)(LUT_D - 1);
        float t3 = fminf(fmaxf(c3, 0.0f), 1.0f) * (float)(LUT_D - 1);
        int i0 = (int)t0; i0 = i0 > LUT_D - 2 ? LUT_D - 2 : i0;
        int i1 = (int)t1; i1 = i1 > LUT_D - 2 ? LUT_D - 2 : i1;
        int i2 = (int)t2; i2 = i2 > LUT_D - 2 ? LUT_D - 2 : i2;
        int i3 = (int)t3; i3 = i3 > LUT_D - 2 ? LUT_D - 2 : i3;
        float f0 = t0 - (float)i0;
        float f1 = t1 - (float)i1;
        float f2 = t2 - (float)i2;
        float f3 = t3 - (float)i3;
        const int base = i0 * 729 + i1 * 81 + i2 * 9 + i3;
        const float g3 = 1.0f - f3;

        float acc = 0.0f;
#pragma unroll
        for (int c0i = 0; c0i < 2; ++c0i) {
            const float w0 = c0i ? f0 : (1.0f - f0);
            const int   o0 = base + c0i * 729;
            float a1 = 0.0f;
#pragma unroll
            for (int c1i = 0; c1i < 2; ++c1i) {
                const float w1 = c1i ? f1 : (1.0f - f1);
                const int   o1 = o0 + c1i * 81;
                float a2 = 0.0f;
#pragma unroll
                for (int c2i = 0; c2i < 2; ++c2i) {
                    const float w2 = c2i ? f2 : (1.0f - f2);
                    const int   o2 = o1 + c2i * 9;
                    const float lo = sLUT[o2];
                    const float hi = sLUT[o2 + 1];
                    a2 += w2 * (lo * g3 + hi * f3);
                }
                a1 += w1 * a2;
            }
            acc += w0 * a1;
        }
        return acc;
    };

    for (int it = tid; it < vecPerBlk; it += TPB) {
        const int p4 = vbase + it;
        const float4 a = X0[p4];
        const float4 bb = X1[p4];
        const float4 c = X2[p4];
        const float4 d = X3[p4];

        float4 r;
        r.x = interp1(a.x, bb.x, c.x, d.x);
        r.y = interp1(a.y, bb.y, c.y, d.y);
        r.z = interp1(a.z, bb.z, c.z, d.z);
        r.w = interp1(a.w, bb.w, c.w, d.w);
        O[p4] = r;
    }
}

extern "C" void kernel_launch(void* const* d_in, const int* in_sizes, int n_in,
                              void* d_out, int out_size, void* d_ws, size_t ws_size,
                              hipStream_t stream) {
    const float* x   = (const float*)d_in[0];
    const float* lut = (const float*)d_in[1];
    float* out       = (float*)d_out;

    dim3 grid(NBLK, NBATCH, 1);
    dim3 block(TPB, 1, 1);
    lut4d_interp_kernel<<<grid, block, 0, stream>>>(x, lut, out);
}